// GraphNeuralConsensus_47777216200866
// MI455X (gfx1250) — compile-verified
//
#include <hip/hip_runtime.h>

#define N_NODES 16384
#define D_DIM   32
#define H_DIM   128
#define ROWS    16            // output rows per block
#define KT      256           // K tile per chunk
#define ADJ_S   (KT + 4)      // 260 floats: 1040B rows, conflict-free A frags (TDM pad 4dw/256dw)
#define MSG_S   40            // 160B rows (16B aligned), lane halves in disjoint banks
#define ACC_S   48            // [agg0 | agg1 | degree]
#define COMB_S  68
#define HID_S   132
#define NCHUNK  (N_NODES / KT)

typedef float v2f __attribute__((ext_vector_type(2)));
typedef float v8f __attribute__((ext_vector_type(8)));
typedef int   v4i __attribute__((vector_size(16)));      // matches async builtin param
typedef unsigned int u32x4 __attribute__((vector_size(16)));
typedef int   i32x8 __attribute__((vector_size(32)));
typedef int   i32x4 __attribute__((vector_size(16)));

#if __has_builtin(__builtin_amdgcn_global_load_async_to_lds_b128)
#define HAVE_ASYNC_LDS 1
#else
#define HAVE_ASYNC_LDS 0
#endif

#if __has_builtin(__builtin_amdgcn_tensor_load_to_lds)
#define HAVE_TDM 1
#else
#define HAVE_TDM 0
#endif

// async ops issued per thread per chunk (messages tile only when TDM handles adj)
#if HAVE_TDM
#define ASYNC_PER_CHUNK 16
#else
#define ASYNC_PER_CHUNK 24
#endif

__device__ __forceinline__ void async_copy16(const float* __restrict__ g, float* l) {
#if HAVE_ASYNC_LDS
  __builtin_amdgcn_global_load_async_to_lds_b128(
      (__attribute__((address_space(1))) v4i*)(g),
      (__attribute__((address_space(3))) v4i*)(l), 0, 0);
#else
  *reinterpret_cast<float4*>(l) = *reinterpret_cast<const float4*>(g);
#endif
}

template <int N>
__device__ __forceinline__ void wait_asynccnt() {
#if __has_builtin(__builtin_amdgcn_s_wait_asynccnt)
  __builtin_amdgcn_s_wait_asynccnt(N);
#else
  asm volatile("s_wait_asynccnt %0" ::"i"(N));
#endif
}

template <int N>
__device__ __forceinline__ void wait_tensorcnt() {
#if __has_builtin(__builtin_amdgcn_s_wait_tensorcnt)
  __builtin_amdgcn_s_wait_tensorcnt(N);
#else
  asm volatile("s_wait_tensorcnt %0" ::"i"(N));
#endif
}

#if HAVE_TDM
// DMA one 16x256-dword 2D tile of adj (row stride N_NODES dwords) into LDS at
// byte offset `ldsOff`, with hardware padding of 4 dwords every 256 dwords so
// the LDS row stride is ADJ_S=260 dwords. Issue from ONE wave only.
__device__ __forceinline__ void tdm_load_adj_tile(const float* gtile, unsigned ldsOff) {
  const unsigned long long ga = (unsigned long long)(size_t)gtile;
  u32x4 g0;
  g0[0] = 1u;                                            // count=1 (valid user D#)
  g0[1] = ldsOff;                                        // lds_addr (bytes)
  g0[2] = (unsigned)(ga & 0xFFFFFFFFu);                  // global_addr[31:0]
  g0[3] = (unsigned)((ga >> 32) & 0x01FFFFFFu) | (2u << 30);  // addr[56:32], type=2
  i32x8 g1;
  g1[0] = (int)((2u << 16)    // data_size = 4 bytes
              | (1u << 20)    // pad_enable
              | (7u << 22)    // pad_interval: 256 dwords
              | (3u << 25));  // pad_amount: 4 dwords
  g1[1] = (int)(((unsigned)N_NODES & 0xFFFFu) << 16);    // tensor_dim0 lo16 -> [31:16]
  g1[2] = (int)((((unsigned)N_NODES >> 16) & 0xFFFFu)    // tensor_dim0 hi16
              | (((unsigned)N_NODES & 0xFFFFu) << 16));  // tensor_dim1 lo16
  g1[3] = (int)((((unsigned)N_NODES >> 16) & 0xFFFFu)    // tensor_dim1 hi16
              | ((unsigned)KT << 16));                   // tile_dim0 = 256
  g1[4] = ROWS;                                          // tile_dim1 = 16
  g1[5] = N_NODES;                                       // tensor_dim0_stride lo32
  g1[6] = 0;
  g1[7] = 0;
  i32x4 z4 = {0, 0, 0, 0};
  i32x8 z8 = {0, 0, 0, 0, 0, 0, 0, 0};
  __builtin_amdgcn_tensor_load_to_lds(g0, g1, z4, z4, z8, 0);  // 6-arg toolchain form
}
#endif

// ---------------- Kernel 1: messages = relu(states@mW1+mb1)@mW2+mb2 ----------------
__global__ __launch_bounds__(128) void gnn_msg_mlp(
    const float* __restrict__ states, const float* __restrict__ mW1,
    const float* __restrict__ mb1,    const float* __restrict__ mW2,
    const float* __restrict__ mb2,    float* __restrict__ messages) {
  __shared__ float st[ROWS * 36];
  __shared__ float hid[ROWS * HID_S];
  const int tid = threadIdx.x;
  const int rowBase = blockIdx.x * ROWS;

  {  // load 16x32 states tile, one float4 per thread (coalesced)
    int r = tid >> 3, c4 = tid & 7;
    float4 v = *reinterpret_cast<const float4*>(states + (size_t)(rowBase + r) * D_DIM + c4 * 4);
    *reinterpret_cast<float4*>(&st[r * 36 + c4 * 4]) = v;
  }
  __syncthreads();

  {  // stage 1: thread tid owns hidden column h = tid
    const int h = tid;
    float acc[ROWS];
    const float b = mb1[h];
#pragma unroll
    for (int r = 0; r < ROWS; ++r) acc[r] = b;
    for (int j = 0; j < D_DIM; ++j) {
      const float w = mW1[j * H_DIM + h];
#pragma unroll
      for (int r = 0; r < ROWS; ++r) acc[r] += st[r * 36 + j] * w;
    }
#pragma unroll
    for (int r = 0; r < ROWS; ++r) hid[r * HID_S + h] = fmaxf(acc[r], 0.0f);
  }
  __syncthreads();

  {  // stage 2: thread owns (rows r0+4i, col c)
    const int c = tid & 31, r0 = tid >> 5;
    float acc[4];
    const float b = mb2[c];
#pragma unroll
    for (int i = 0; i < 4; ++i) acc[i] = b;
    for (int j = 0; j < H_DIM; ++j) {
      const float w = mW2[j * D_DIM + c];
#pragma unroll
      for (int i = 0; i < 4; ++i) acc[i] += hid[(r0 + 4 * i) * HID_S + j] * w;
    }
#pragma unroll
    for (int i = 0; i < 4; ++i)
      messages[(size_t)(rowBase + r0 + 4 * i) * D_DIM + c] = acc[i];
  }
}

// ---- Kernel 2: fused (adj row-sum + adj@messages) via WMMA f32, then update MLP ----
__global__ __launch_bounds__(128) void gnn_aggregate_update(
    const float* __restrict__ states, const float* __restrict__ adj,
    const float* __restrict__ messages,
    const float* __restrict__ uW1, const float* __restrict__ ub1,
    const float* __restrict__ uW2, const float* __restrict__ ub2,
    float* __restrict__ out) {
  __shared__ float sAdj[2][ROWS * ADJ_S];   // double-buffered adj tile (TDM-filled)
  __shared__ float sMsg[2][KT * MSG_S];     // double-buffered messages tile (async-filled)
  __shared__ float sAcc[ROWS * ACC_S];
  __shared__ float sComb[ROWS * COMB_S];
  __shared__ float sHid[ROWS * HID_S];

  const int tid = threadIdx.x;
  const int lane = tid & 31;
  // force wave id scalar so the K-step loop compiles as an SALU loop (EXEC untouched)
  const int wave = __builtin_amdgcn_readfirstlane(tid >> 5);
  const int rowBase = blockIdx.x * ROWS;

  for (int i = tid; i < ROWS * ACC_S; i += 128) sAcc[i] = 0.0f;

  v8f c0 = {0, 0, 0, 0, 0, 0, 0, 0};  // aggregated cols 0..15
  v8f c1 = {0, 0, 0, 0, 0, 0, 0, 0};  // aggregated cols 16..31
  v8f cd = {0, 0, 0, 0, 0, 0, 0, 0};  // degree in column 0

  // constant B fragment: 4x16 matrix with column 0 == 1 (fused row-sum)
  v2f bdeg;
  bdeg.x = ((lane & 15) == 0) ? 1.0f : 0.0f;
  bdeg.y = bdeg.x;

  // A frag (16x4 f32): lanes 0-15 -> M=lane, K=kk+0/1 ; lanes 16-31 -> M=lane-16, K=kk+2/3
  const int aRow = lane & 15;
  const int aK   = (lane >> 4) << 1;
  // B frag (4x16 f32): lower lanes hold K=kk+0/1, upper lanes K=kk+2/3; N = lane&15
  const int bRowOff = (lane >> 4) << 1;
  const int bCol    = lane & 15;

  // ---- stage adj tile for `chunk` into buffer `buf` ----
  auto issue_adj = [&](int chunk, int buf) {
    const int kBase = chunk * KT;
#if HAVE_TDM
    if (wave == 0) {
      const unsigned ldsOff =
          (unsigned)(size_t)(__attribute__((address_space(3))) char*)&sAdj[buf][0];
      tdm_load_adj_tile(adj + (size_t)rowBase * N_NODES + kBase, ldsOff);
    }
#else
#pragma unroll
    for (int u = 0; u < 8; ++u) {            // adj: 16x256 floats, 8 x b128 per thread
      const int s = tid + u * 128;
      const int r = s >> 6, c4 = s & 63;
      async_copy16(adj + (size_t)(rowBase + r) * N_NODES + kBase + c4 * 4,
                   &sAdj[buf][r * ADJ_S + c4 * 4]);
    }
#endif
  };
  // ---- stage messages tile for `chunk` into buffer `buf` (async, all waves) ----
  auto issue_msg = [&](int chunk, int buf) {
    const int kBase = chunk * KT;
#pragma unroll
    for (int u = 0; u < 16; ++u) {           // msg: 256x32 floats, 16 x b128 per thread
      const int s = tid + u * 128;
      const int kr = s >> 3, c4 = s & 7;
      async_copy16(messages + (size_t)(kBase + kr) * D_DIM + c4 * 4,
                   &sMsg[buf][kr * MSG_S + c4 * 4]);
    }
  };

  issue_adj(0, 0);                           // prologue: fill buffer 0
  issue_msg(0, 0);

  for (int chunk = 0; chunk < NCHUNK; ++chunk) {
    const int buf = chunk & 1;
    if (chunk + 1 < NCHUNK) {
      issue_adj(chunk + 1, buf ^ 1);         // prefetch next chunk into other buffer
      issue_msg(chunk + 1, buf ^ 1);
      wait_asynccnt<ASYNC_PER_CHUNK>();      // in-order: current chunk's msg loads done
#if HAVE_TDM
      if (wave == 0) wait_tensorcnt<1>();    // current chunk's TDM done (1 in flight)
#endif
    } else {
      wait_asynccnt<0>();
#if HAVE_TDM
      if (wave == 0) wait_tensorcnt<0>();
#endif
    }
    __syncthreads();

    // 4 waves split the 64 K-steps of this chunk (uniform scalar loop)
    const float* __restrict__ sa = sAdj[buf];
    const float* __restrict__ sm = sMsg[buf];
    for (int kk = wave * 4; kk < KT; kk += 16) {
      v2f a, b0, b1;
      const float* ap = &sa[aRow * ADJ_S + kk + aK];
      a.x = ap[0];  a.y = ap[1];
      const int r0 = kk + bRowOff;
      b0.x = sm[r0 * MSG_S + bCol];
      b0.y = sm[(r0 + 1) * MSG_S + bCol];
      b1.x = sm[r0 * MSG_S + 16 + bCol];
      b1.y = sm[(r0 + 1) * MSG_S + 16 + bCol];
      c0 = __builtin_amdgcn_wmma_f32_16x16x4_f32(false, a, false, b0,   (short)0, c0, false, false);
      c1 = __builtin_amdgcn_wmma_f32_16x16x4_f32(false, a, false, b1,   (short)0, c1, false, false);
      cd = __builtin_amdgcn_wmma_f32_16x16x4_f32(false, a, false, bdeg, (short)0, cd, false, false);
    }
    __syncthreads();   // all waves done with buf before it is refilled next iteration
  }

  // cross-wave reduction of C fragments (C layout: VGPR j -> rows j / j+8)
  {
    const int M0 = (lane & 16) ? 8 : 0;
    const int col = lane & 15;
#pragma unroll
    for (int vg = 0; vg < 8; ++vg) {
      const int row = M0 + vg;
      atomicAdd(&sAcc[row * ACC_S + col],      c0[vg]);
      atomicAdd(&sAcc[row * ACC_S + 16 + col], c1[vg]);
      atomicAdd(&sAcc[row * ACC_S + 32 + col], cd[vg]);
    }
  }
  __syncthreads();

  // combined = [states | aggregated/max(degree,1)]
  for (int e = tid; e < ROWS * 2 * D_DIM; e += 128) {
    const int r = e >> 6, c = e & 63;
    float v;
    if (c < D_DIM) {
      v = states[(size_t)(rowBase + r) * D_DIM + c];
    } else {
      const float deg = fmaxf(sAcc[r * ACC_S + 32], 1.0f);
      v = sAcc[r * ACC_S + (c - D_DIM)] / deg;
    }
    sComb[r * COMB_S + c] = v;
  }
  __syncthreads();

  // update MLP stage 1: hid = relu(combined @ uW1 + ub1)
  {
    const int h = tid;
    float acc[ROWS];
    const float b = ub1[h];
#pragma unroll
    for (int r = 0; r < ROWS; ++r) acc[r] = b;
    for (int j = 0; j < 2 * D_DIM; ++j) {
      const float w = uW1[j * H_DIM + h];
#pragma unroll
      for (int r = 0; r < ROWS; ++r) acc[r] += sComb[r * COMB_S + j] * w;
    }
#pragma unroll
    for (int r = 0; r < ROWS; ++r) sHid[r * HID_S + h] = fmaxf(acc[r], 0.0f);
  }
  __syncthreads();

  // update MLP stage 2: out = hid @ uW2 + ub2
  {
    const int c = tid & 31, r0 = tid >> 5;
    float acc[4];
    const float b = ub2[c];
#pragma unroll
    for (int i = 0; i < 4; ++i) acc[i] = b;
    for (int j = 0; j < H_DIM; ++j) {
      const float w = uW2[j * D_DIM + c];
#pragma unroll
      for (int i = 0; i < 4; ++i) acc[i] += sHid[(r0 + 4 * i) * HID_S + j] * w;
    }
#pragma unroll
    for (int i = 0; i < 4; ++i)
      out[(size_t)(rowBase + r0 + 4 * i) * D_DIM + c] = acc[i];
  }
}

extern "C" void kernel_launch(void* const* d_in, const int* in_sizes, int n_in,
                              void* d_out, int out_size, void* d_ws, size_t ws_size,
                              hipStream_t stream) {
  const float* states = (const float*)d_in[0];
  const float* adj    = (const float*)d_in[1];
  const float* mW1    = (const float*)d_in[2];
  const float* mb1    = (const float*)d_in[3];
  const float* mW2    = (const float*)d_in[4];
  const float* mb2    = (const float*)d_in[5];
  const float* uW1    = (const float*)d_in[6];
  const float* ub1    = (const float*)d_in[7];
  const float* uW2    = (const float*)d_in[8];
  const float* ub2    = (const float*)d_in[9];
  float* out  = (float*)d_out;
  float* msgs = (float*)d_ws;   // N*D fp32 = 2 MB scratch for messages

  dim3 grid(N_NODES / ROWS);    // 1024 blocks, 16 rows each
  gnn_msg_mlp<<<grid, 128, 0, stream>>>(states, mW1, mb1, mW2, mb2, msgs);
  gnn_aggregate_update<<<grid, 128, 0, stream>>>(states, adj, msgs,
                                                 uW1, ub1, uW2, ub2, out);
}